// self_similarity_layer_8229157339490
// MI455X (gfx1250) — compile-verified
//
#include <hip/hip_runtime.h>
#include <hip/hip_bf16.h>

typedef float v2f __attribute__((ext_vector_type(2)));
typedef float v8f __attribute__((ext_vector_type(8)));

#define B_     8
#define H_     64
#define W_     64
#define C_     256
#define OH     65
#define OW     65
#define NPIX   (B_ * H_ * W_)          // 32768
#define NOUT   (B_ * OH * OW)          // 33800
#define LDSTR  68                      // padded LDS row stride (floats): 16B aligned, bank-spread
#define CCHUNK 64                      // channels staged per LDS chunk
#define BUFELTS (H_ * LDSTR)           // floats per staging buffer

// ---------------------------------------------------------------------------
// CDNA5 async global->LDS copy (ASYNCcnt-tracked, no VGPR round trip)
// ---------------------------------------------------------------------------
__device__ __forceinline__ void async_b128(unsigned lds_off, const float* g) {
    asm volatile("global_load_async_to_lds_b128 %0, %1, off"
                 :: "v"(lds_off), "v"(g) : "memory");
}
__device__ __forceinline__ void wait_async4() {
    asm volatile("s_wait_asynccnt 0x4" ::: "memory");
}
__device__ __forceinline__ void wait_async0() {
    asm volatile("s_wait_asynccnt 0x0" ::: "memory");
}

// ---------------------------------------------------------------------------
// Kernel 1: S[b,u,v] = sum_c X[b,u,v,c]^2
// ---------------------------------------------------------------------------
__global__ __launch_bounds__(256) void k_sumsq(const float* __restrict__ X,
                                               float* __restrict__ S) {
    int idx = blockIdx.x * blockDim.x + threadIdx.x;
    if (idx >= NPIX) return;
    const float* p = X + (size_t)idx * C_;
    float acc = 0.f;
#pragma unroll 4
    for (int c = 0; c < C_; c += 4) {
        float4 v = *(const float4*)(p + c);
        acc += v.x * v.x + v.y * v.y + v.z * v.z + v.w * v.w;
    }
    S[idx] = acc;
}

// ---------------------------------------------------------------------------
// Kernel 2: per-batch 2D inclusive prefix sum (integral image), 64x64
// ---------------------------------------------------------------------------
__global__ __launch_bounds__(64) void k_integral(const float* __restrict__ S,
                                                 float* __restrict__ I) {
    __shared__ float tile[H_][W_ + 1];
    int b = blockIdx.x;
    int t = threadIdx.x;
    for (int r = 0; r < H_; ++r) tile[r][t] = S[(b * H_ + r) * W_ + t];
    __syncthreads();
    float run = 0.f;
    for (int c = 0; c < W_; ++c) { run += tile[t][c]; tile[t][c] = run; }
    __syncthreads();
    run = 0.f;
    for (int r = 0; r < H_; ++r) { run += tile[r][t]; tile[r][t] = run; }
    __syncthreads();
    for (int r = 0; r < H_; ++r) I[(b * H_ + r) * W_ + t] = tile[r][t];
}

// ---------------------------------------------------------------------------
// Kernel 3 (WMMA core): cross-correlation ab[b, 32+dy, 32+dx]
// One workgroup per (b, dy), 512 thr = 16 wave32. Each wave owns a 32x32
// quad (2x2 WMMA tiles -> 4 accumulators, 1 ds_load_b64 per WMMA) and a
// 16-channel K-slice of each 64-channel chunk (K-partials merge in the
// ds_add_f32 diagonal reduction). Chunks are staged with double-buffered
// GLOBAL_LOAD_ASYNC_TO_LDS_B128 so copy of chunk i+1 overlaps WMMA on i.
// ---------------------------------------------------------------------------
__global__ __launch_bounds__(512) void k_ab_wmma(const float* __restrict__ X,
                                                 float* __restrict__ ab) {
    __shared__ float ldsA[2][BUFELTS];
    __shared__ float ldsB[2][BUFELTS];
    __shared__ float abacc[OW];

    const int b    = blockIdx.x / OH;
    const int dyi  = blockIdx.x % OH;    // 0..64
    const int dy   = dyi - 32;           // -32..32
    const int tid  = threadIdx.x;
    const int wave = tid >> 5;           // 0..15
    const int lane = tid & 31;
    const int qm   = (wave & 3) >> 1;    // quad row (v0 block of 32)
    const int qn   = wave & 1;           // quad col (vb block of 32)
    const int ksl  = wave >> 2;          // K-slice 0..3 (16 channels each)

    if (tid < OW) abacc[tid] = 0.f;

    const int u0lo = (dy < 0) ? -dy : 0;
    const int u0hi = (dy < 0) ? H_ : H_ - dy;
    const int nch  = (u0hi - u0lo) * (C_ / CCHUNK);   // >= 128 chunks

    const float* X0 = X;                               // batch 0 (filter)
    const float* Xb = X + (size_t)b * H_ * W_ * C_;

    const unsigned offA = (unsigned)(uintptr_t)&ldsA[0][0];
    const unsigned offB = (unsigned)(uintptr_t)&ldsB[0][0];

    // stage chunk ci into buffer `buf`: 64 rows x 64 c floats per matrix,
    // 1024 b128 transfers per matrix, 2 per thread -> 4 async ops per wave
    auto stage = [&](int ci, int buf) {
        const int u0 = u0lo + (ci >> 2);
        const int cb = (ci & 3) << 6;
        const float* rowA = X0 + (size_t)u0 * W_ * C_ + cb;
        const float* rowB = Xb + (size_t)(u0 + dy) * W_ * C_ + cb;
        const unsigned bA = offA + (unsigned)buf * (BUFELTS * 4);
        const unsigned bB = offB + (unsigned)buf * (BUFELTS * 4);
#pragma unroll
        for (int r = 0; r < 2; ++r) {
            int f  = tid * 2 + r;                      // 0..1023
            int v  = f >> 4;                           // row 0..63
            int c4 = (f & 15) << 2;                    // 0,4,...,60
            unsigned loff = (unsigned)(v * LDSTR + c4) * 4u;
            async_b128(bA + loff, rowA + v * C_ + c4);
            async_b128(bB + loff, rowB + v * C_ + c4);
        }
    };

    // fragment addressing (A 16x4 / B 4x16 f32 layouts): lanes 0-15 hold
    // K0/K1 in the two operand VGPRs, lanes 16-31 hold K2/K3; M(N)=lane%16
    const int khalf = (lane >> 4) << 1;                // 0 or 2
    const int fA    = (qm * 32 + (lane & 15)) * LDSTR + ksl * 16 + khalf;
    const int fB    = (qn * 32 + (lane & 15)) * LDSTR + ksl * 16 + khalf;

    v8f a00 = {}, a01 = {}, a10 = {}, a11 = {};

    stage(0, 0);
    for (int ci = 0; ci < nch; ++ci) {
        const bool hasnext = (ci + 1) < nch;
        if (hasnext) { stage(ci + 1, (ci + 1) & 1); wait_async4(); }
        else         { wait_async0(); }
        __syncthreads();                               // chunk ci visible

        const float* pA = &ldsA[ci & 1][fA];
        const float* pB = &ldsB[ci & 1][fB];
#pragma unroll
        for (int j = 0; j < 4; ++j) {                  // 4 k-steps of K=4
            v2f av0 = *(const v2f*)(pA + j * 4);
            v2f av1 = *(const v2f*)(pA + 16 * LDSTR + j * 4);
            v2f bv0 = *(const v2f*)(pB + j * 4);
            v2f bv1 = *(const v2f*)(pB + 16 * LDSTR + j * 4);
            a00 = __builtin_amdgcn_wmma_f32_16x16x4_f32(false, av0, false, bv0,
                                                        (short)0, a00, false, false);
            a01 = __builtin_amdgcn_wmma_f32_16x16x4_f32(false, av0, false, bv1,
                                                        (short)0, a01, false, false);
            a10 = __builtin_amdgcn_wmma_f32_16x16x4_f32(false, av1, false, bv0,
                                                        (short)0, a10, false, false);
            a11 = __builtin_amdgcn_wmma_f32_16x16x4_f32(false, av1, false, bv1,
                                                        (short)0, a11, false, false);
        }
        __syncthreads();                               // done reading buf[ci&1]
    }

    // scatter quad diagonals into the 65 dx bins (C/D layout: VGPR r ->
    // M = r + (lane>=16 ? 8:0), N = lane%16)
    const int mb = qm * 32 + ((lane >> 4) << 3);
    const int nb = qn * 32 + (lane & 15);
#pragma unroll
    for (int r = 0; r < 8; ++r) {
        int dx;
        dx = (nb)      - (mb + r);
        if (dx >= -32 && dx <= 32) atomicAdd(&abacc[dx + 32], a00[r]);
        dx = (nb + 16) - (mb + r);
        if (dx >= -32 && dx <= 32) atomicAdd(&abacc[dx + 32], a01[r]);
        dx = (nb)      - (mb + 16 + r);
        if (dx >= -32 && dx <= 32) atomicAdd(&abacc[dx + 32], a10[r]);
        dx = (nb + 16) - (mb + 16 + r);
        if (dx >= -32 && dx <= 32) atomicAdd(&abacc[dx + 32], a11[r]);
    }
    __syncthreads();
    if (tid < OW) ab[((size_t)b * OH + dyi) * OW + tid] = abacc[tid];
}

// ---------------------------------------------------------------------------
// Kernel 4: a^2 and b^2 via integral-image rectangle queries
// ---------------------------------------------------------------------------
__device__ __forceinline__ float iat(const float* I, int r, int c) {
    return (r < 0 || c < 0) ? 0.f : I[r * W_ + c];
}
__device__ __forceinline__ float rectsum(const float* I, int r0, int r1,
                                         int c0, int c1) {
    return iat(I, r1 - 1, c1 - 1) - iat(I, r0 - 1, c1 - 1)
         - iat(I, r1 - 1, c0 - 1) + iat(I, r0 - 1, c0 - 1);
}

__global__ __launch_bounds__(256) void k_a2b2(const float* __restrict__ I,
                                              float* __restrict__ a2,
                                              float* __restrict__ b2) {
    int idx = blockIdx.x * blockDim.x + threadIdx.x;
    if (idx >= NOUT) return;
    int b = idx / (OH * OW);
    int rem = idx % (OH * OW);
    int y = rem / OW, x = rem % OW;
    const float* Ib = I + (size_t)b * H_ * W_;
    int r0 = max(0, y - 32), r1 = min(H_, y + 32);
    int c0 = max(0, x - 32), c1 = min(W_, x + 32);
    a2[idx] = rectsum(Ib, r0, r1, c0, c1);
    if (b == 0) {
        int br0 = max(0, 32 - y), br1 = min(H_, 96 - y);
        int bc0 = max(0, 32 - x), bc1 = min(W_, 96 - x);
        b2[rem] = rectsum(I, br0, br1, bc0, bc1);
    }
}

// ---------------------------------------------------------------------------
// Kernel 5: sim -> conv3x3(1->8) + ReLU -> conv3x3(8->1), SAME (zero pad)
// ---------------------------------------------------------------------------
__global__ __launch_bounds__(256) void k_head(const float* __restrict__ a2,
                                              const float* __restrict__ b2,
                                              const float* __restrict__ ab,
                                              const float* __restrict__ w1,
                                              const float* __restrict__ b1,
                                              const float* __restrict__ w2,
                                              const float* __restrict__ b2bias,
                                              float* __restrict__ out) {
    int idx = blockIdx.x * blockDim.x + threadIdx.x;
    if (idx >= NOUT) return;
    int b = idx / (OH * OW);
    int rem = idx % (OH * OW);
    int y = rem / OW, x = rem % OW;

    float outv = b2bias[0];
#pragma unroll
    for (int qi = 0; qi < 3; ++qi) {
#pragma unroll
        for (int qj = 0; qj < 3; ++qj) {
            int hy = y + qi - 1, hx = x + qj - 1;
            if (hy < 0 || hy >= OH || hx < 0 || hx >= OW) continue;
            float h[8];
#pragma unroll
            for (int o = 0; o < 8; ++o) h[o] = b1[o];
#pragma unroll
            for (int pi = 0; pi < 3; ++pi) {
#pragma unroll
                for (int pj = 0; pj < 3; ++pj) {
                    int sy = hy + pi - 1, sx = hx + pj - 1;
                    if (sy < 0 || sy >= OH || sx < 0 || sx >= OW) continue;
                    int o2 = (b * OH + sy) * OW + sx;
                    float av = a2[o2];
                    float s  = -((av + b2[sy * OW + sx] - 2.f * ab[o2]) / av);
                    const float* w1p = w1 + (pi * 3 + pj) * 8;   // (3,3,1,8)
#pragma unroll
                    for (int o = 0; o < 8; ++o) h[o] = fmaf(s, w1p[o], h[o]);
                }
            }
            const float* w2p = w2 + (qi * 3 + qj) * 8;           // (3,3,8,1)
#pragma unroll
            for (int o = 0; o < 8; ++o)
                outv = fmaf(fmaxf(h[o], 0.f), w2p[o], outv);
        }
    }
    out[idx] = outv;
}

// ---------------------------------------------------------------------------
extern "C" void kernel_launch(void* const* d_in, const int* in_sizes, int n_in,
                              void* d_out, int out_size, void* d_ws, size_t ws_size,
                              hipStream_t stream) {
    const float* X   = (const float*)d_in[0];  // [8,64,64,256]
    const float* w1  = (const float*)d_in[1];  // [3,3,1,8]
    const float* b1  = (const float*)d_in[2];  // [8]
    const float* w2  = (const float*)d_in[3];  // [3,3,8,1]
    const float* b2w = (const float*)d_in[4];  // [1]
    float* out = (float*)d_out;                // [8,65,65,1]

    float* S  = (float*)d_ws;                  // 32768
    float* I  = S  + NPIX;                     // 32768
    float* ab = I  + NPIX;                     // 33800
    float* a2 = ab + NOUT;                     // 33800
    float* b2 = a2 + NOUT;                     // 4225

    k_sumsq<<<(NPIX + 255) / 256, 256, 0, stream>>>(X, S);
    k_integral<<<B_, 64, 0, stream>>>(S, I);
    k_ab_wmma<<<B_ * OH, 512, 0, stream>>>(X, ab);
    k_a2b2<<<(NOUT + 255) / 256, 256, 0, stream>>>(I, a2, b2);
    k_head<<<(NOUT + 255) / 256, 256, 0, stream>>>(a2, b2, ab, w1, b1, w2, b2w, out);
}